// AgentAndNode_embedding_76295799046657
// MI455X (gfx1250) — compile-verified
//
#include <hip/hip_runtime.h>

#define NT_TOTAL 128000
#define HID 64
#define DEG 16
#define N_AGENT 10
#define N_NODES 1000
#define N_BATCH 128
#define F_C_DIM ((1 + N_AGENT) * HID) /* 704 */
#define BN_EPS 1e-5f
#define N_FREE (N_NODES - N_AGENT)    /* 990 */

typedef float v2f __attribute__((ext_vector_type(2)));
typedef float v4f __attribute__((ext_vector_type(4)));
typedef float v8f __attribute__((ext_vector_type(8)));

// ---------- layer-0 padding ----------
__global__ __launch_bounds__(256) void pad_x_kernel(const float* __restrict__ x,
                                                    float* __restrict__ h) {
  long idx = (long)blockIdx.x * 256 + threadIdx.x;
  if (idx >= (long)NT_TOTAL * HID) return;
  int c = (int)(idx & 63);
  long row = idx >> 6;
  h[idx] = (c < 2) ? x[row * 2 + c] : 0.0f;
}

__global__ __launch_bounds__(256) void pad_w1_kernel(const float* __restrict__ W1,
                                                     float* __restrict__ W1p) {
  int idx = blockIdx.x * 256 + threadIdx.x;
  if (idx >= 64 * 64) return;
  int r = idx >> 6;
  W1p[idx] = (r < 2) ? W1[idx] : 0.0f;
}

// ---------- GIN aggregation: m[i] = h[i] + sum_{j<16} h[src[16i+j]] ----------
// dst(e) == e/16, so no atomics; one wave per node, float2 per lane (64 ch).
__global__ __launch_bounds__(256) void agg_kernel(const float* __restrict__ h,
                                                  const int* __restrict__ src,
                                                  float* __restrict__ m) {
  int node = (int)((blockIdx.x * 256 + threadIdx.x) >> 5);
  int lane = threadIdx.x & 31;
  if (node >= NT_TOTAL) return;
  const float2* hp = (const float2*)h;
  long self = (long)node * 32 + lane;
  float2 v = hp[self];
  float sx = v.x, sy = v.y;
  const int* s = src + (long)node * DEG;
#pragma unroll
  for (int j = 0; j < DEG; ++j) {
    long r = (long)s[j];
    float2 u = hp[r * 32 + lane];
    sx += u.x;
    sy += u.y;
  }
  float2 o; o.x = sx; o.y = sy;
  ((float2*)m)[self] = o;
}

// ---------- fused 2-GEMM MLP via V_WMMA_F32_16X16X4_F32 ----------
// m (in-place) <- relu(m @ W1 + b1) @ W2 + b2 ; 4 waves/block, 16 rows/wave.
__global__ __launch_bounds__(128) void mlp_kernel(float* __restrict__ m,
                                                  const float* __restrict__ W1,
                                                  const float* __restrict__ b1,
                                                  const float* __restrict__ W2,
                                                  const float* __restrict__ b2) {
  __shared__ float sW1[64 * 64];
  __shared__ float sW2[64 * 64];
  __shared__ float sB1[64];
  __shared__ float sB2[64];
  __shared__ float sT[64 * 64];  // 64-row tile: input, then intermediate t

  int tid = threadIdx.x;
  for (int i = tid; i < 64 * 64; i += 128) { sW1[i] = W1[i]; sW2[i] = W2[i]; }
  if (tid < 64) { sB1[tid] = b1[tid]; sB2[tid] = b2[tid]; }
  long base = (long)blockIdx.x * 64 * 64;
  const v4f* gm = (const v4f*)(m + base);
  v4f* smv = (v4f*)sT;
  for (int i = tid; i < 64 * 64 / 4; i += 128) smv[i] = gm[i];
  __syncthreads();

  int wave = tid >> 5;
  int lane = tid & 31;
  int ln = lane & 15;                   // row (A) / col (B,C)
  int halfsel = (lane < 16) ? 0 : 2;    // K sub-slice per lane half
  int rowsel = (lane < 16) ? 0 : 8;     // C/D row offset per lane half
  const float* Arow = &sT[(wave * 16 + ln) * 64];

  // ---- GEMM1: t = relu(m @ W1 + b1) ----
  v8f acc[4] = {};
  for (int kk = 0; kk < 16; ++kk) {
    int k0 = kk * 4 + halfsel;
    v2f a;
    a[0] = Arow[k0];
    a[1] = Arow[k0 + 1];
#pragma unroll
    for (int nt = 0; nt < 4; ++nt) {
      v2f bfr;
      bfr[0] = sW1[k0 * 64 + nt * 16 + ln];
      bfr[1] = sW1[(k0 + 1) * 64 + nt * 16 + ln];
      acc[nt] = __builtin_amdgcn_wmma_f32_16x16x4_f32(
          false, a, false, bfr, (short)0, acc[nt], false, false);
    }
  }
#pragma unroll
  for (int nt = 0; nt < 4; ++nt) {
    float bias = sB1[nt * 16 + ln];
#pragma unroll
    for (int vr = 0; vr < 8; ++vr) {
      float t = fmaxf(acc[nt][vr] + bias, 0.0f);
      // wave-private region; same-wave LDS ops are in-order -> no barrier
      sT[(wave * 16 + rowsel + vr) * 64 + nt * 16 + ln] = t;
    }
  }

  // ---- GEMM2: out = t @ W2 + b2 ----
  v8f acc2[4] = {};
  for (int kk = 0; kk < 16; ++kk) {
    int k0 = kk * 4 + halfsel;
    v2f a;
    a[0] = Arow[k0];
    a[1] = Arow[k0 + 1];
#pragma unroll
    for (int nt = 0; nt < 4; ++nt) {
      v2f bfr;
      bfr[0] = sW2[k0 * 64 + nt * 16 + ln];
      bfr[1] = sW2[(k0 + 1) * 64 + nt * 16 + ln];
      acc2[nt] = __builtin_amdgcn_wmma_f32_16x16x4_f32(
          false, a, false, bfr, (short)0, acc2[nt], false, false);
    }
  }
#pragma unroll
  for (int nt = 0; nt < 4; ++nt) {
    float bias = sB2[nt * 16 + ln];
#pragma unroll
    for (int vr = 0; vr < 8; ++vr) {
      m[base + (long)(wave * 16 + rowsel + vr) * 64 + nt * 16 + ln] =
          acc2[nt][vr] + bias;
    }
  }
}

// ---------- batch-norm statistics (sum, sumsq per channel) ----------
__global__ __launch_bounds__(256) void bn_stats_kernel(const float* __restrict__ m,
                                                       float* __restrict__ stats) {
  __shared__ float ssum[64];
  __shared__ float ssq[64];
  int tid = threadIdx.x;
  if (tid < 64) { ssum[tid] = 0.0f; ssq[tid] = 0.0f; }
  __syncthreads();
  long n = (long)NT_TOTAL * HID;
  long stride = (long)gridDim.x * 256;  // multiple of 64 -> channel fixed/thread
  float s = 0.0f, s2 = 0.0f;
  for (long i = (long)blockIdx.x * 256 + tid; i < n; i += stride) {
    float v = m[i];
    s += v;
    s2 += v * v;
  }
  int c = tid & 63;
  atomicAdd(&ssum[c], s);
  atomicAdd(&ssq[c], s2);
  __syncthreads();
  if (tid < 64) {
    atomicAdd(&stats[tid], ssum[tid]);
    atomicAdd(&stats[64 + tid], ssq[tid]);
  }
}

__global__ __launch_bounds__(256) void bn_apply_kernel(float* __restrict__ m,
                                                       const float* __restrict__ stats,
                                                       const float* __restrict__ gamma,
                                                       const float* __restrict__ beta) {
  long idx = (long)blockIdx.x * 256 + threadIdx.x;
  if (idx >= (long)NT_TOTAL * HID) return;
  int c = (int)(idx & 63);
  const float invN = 1.0f / (float)NT_TOTAL;
  float mu = stats[c] * invN;
  float var = fmaxf(stats[64 + c] * invN - mu * mu, 0.0f);
  float inv = rsqrtf(var + BN_EPS);
  float v = gamma[c] * (m[idx] - mu) * inv + beta[c];
  m[idx] = fmaxf(v, 0.0f);
}

// ---------- f_c = [mean over nodes, first 10 node embeddings] ----------
__global__ __launch_bounds__(256) void fc_build_kernel(const float* __restrict__ h,
                                                       float* __restrict__ fc) {
  __shared__ float red[256];
  int b = blockIdx.x, tid = threadIdx.x;
  int c = tid & 63, sub = tid >> 6;
  const float* hb = h + (long)b * N_NODES * HID;
  float s = 0.0f;
  for (int n = sub; n < N_NODES; n += 4) s += hb[(long)n * 64 + c];
  red[tid] = s;
  __syncthreads();
  if (tid < 64)
    fc[(long)b * F_C_DIM + tid] =
        (red[tid] + red[64 + tid] + red[128 + tid] + red[192 + tid]) *
        (1.0f / (float)N_NODES);
  for (int i = tid; i < N_AGENT * HID; i += 256)
    fc[(long)b * F_C_DIM + 64 + i] = hb[i];  // nodes 0..9 are contiguous
}

// ---------- q[b,a,:] = f_c[b] @ Wq[a] + bq[a] (tiny: 115 MFLOP) ----------
__global__ __launch_bounds__(256) void q_kernel(const float* __restrict__ fc,
                                                const float* __restrict__ Wq,
                                                const float* __restrict__ bq,
                                                float* __restrict__ qout) {
  int idx = blockIdx.x * 256 + threadIdx.x;
  if (idx >= N_BATCH * N_AGENT * HID) return;
  int j = idx & 63;
  int a = (idx >> 6) % N_AGENT;
  int b = idx / (N_AGENT * HID);
  const float* f = fc + (long)b * F_C_DIM;
  const float* w = Wq + (long)a * F_C_DIM * 64;
  float s = bq[a * 64 + j];
  for (int i = 0; i < F_C_DIM; ++i) s += f[i] * w[(long)i * 64 + j];
  qout[idx] = s;
}

// ---------- copy f = nodes_h[:, 10:, :] to output ----------
__global__ __launch_bounds__(256) void fcopy_kernel(const float* __restrict__ h,
                                                    float* __restrict__ fout) {
  long idx = (long)blockIdx.x * 256 + threadIdx.x;
  long total = (long)N_BATCH * N_FREE * 64;
  if (idx >= total) return;
  int c = (int)(idx & 63);
  long rf = idx >> 6;
  long b = rf / N_FREE;
  long mm = rf - b * N_FREE;
  fout[idx] = h[((b * N_NODES) + N_AGENT + mm) * 64 + c];
}

// ---------- attention, algebraically collapsed ----------
// u = f·(Wk q) + bk·q ; softmax ; out = (Σ w f)·Wv + bv
__global__ __launch_bounds__(256) void attn_kernel(const float* __restrict__ h,
                                                   const float* __restrict__ q,
                                                   const float* __restrict__ Wk,
                                                   const float* __restrict__ bk,
                                                   const float* __restrict__ Wv,
                                                   const float* __restrict__ bv,
                                                   float* __restrict__ out) {
  __shared__ float rsh[64];
  __shared__ float ssh[64];
  __shared__ float wsh[992];
  __shared__ float red[256];
  __shared__ float csh;
  int b = blockIdx.x, a = blockIdx.y, tid = threadIdx.x;
  const float* qv = q + ((long)b * N_AGENT + a) * 64;
  if (tid < 64) {
    const float* wr = Wk + ((long)a * 64 + tid) * 64;
    float s = 0.0f;
    for (int j = 0; j < 64; ++j) s += wr[j] * qv[j];
    rsh[tid] = s;
  } else if (tid == 64) {
    const float* bkr = bk + (long)a * 64;
    float s = 0.0f;
    for (int j = 0; j < 64; ++j) s += bkr[j] * qv[j];
    csh = s;
  }
  __syncthreads();
  const float* fb = h + ((long)b * N_NODES + N_AGENT) * 64;
  float lmax = -3.0e38f;
  for (int m = tid; m < N_FREE; m += 256) {
    const float* fr = fb + (long)m * 64;
    float s = csh;
    for (int j = 0; j < 64; ++j) s += fr[j] * rsh[j];
    s *= 0.125f;  // 1/sqrt(64)
    wsh[m] = s;
    lmax = fmaxf(lmax, s);
  }
  red[tid] = lmax;
  __syncthreads();
  for (int off = 128; off > 0; off >>= 1) {
    if (tid < off) red[tid] = fmaxf(red[tid], red[tid + off]);
    __syncthreads();
  }
  float mx = red[0];
  __syncthreads();
  float lsum = 0.0f;
  for (int m = tid; m < N_FREE; m += 256) {
    float e = expf(wsh[m] - mx);
    wsh[m] = e;
    lsum += e;
  }
  red[tid] = lsum;
  __syncthreads();
  for (int off = 128; off > 0; off >>= 1) {
    if (tid < off) red[tid] += red[tid + off];
    __syncthreads();
  }
  float invS = 1.0f / red[0];
  __syncthreads();
  int c = tid & 63, sub = tid >> 6;
  float acc = 0.0f;
  for (int m = sub; m < N_FREE; m += 4) acc += wsh[m] * fb[(long)m * 64 + c];
  red[tid] = acc;
  __syncthreads();
  if (tid < 64)
    ssh[tid] = (red[tid] + red[64 + tid] + red[128 + tid] + red[192 + tid]) * invS;
  __syncthreads();
  if (tid < 64) {
    float s = bv[(long)a * 64 + tid];
    for (int j = 0; j < 64; ++j) s += ssh[j] * Wv[((long)a * 64 + j) * 64 + tid];
    out[((long)b * N_AGENT + a) * 64 + tid] = s;
  }
}

extern "C" void kernel_launch(void* const* d_in, const int* in_sizes, int n_in,
                              void* d_out, int out_size, void* d_ws, size_t ws_size,
                              hipStream_t stream) {
  (void)in_sizes; (void)n_in; (void)out_size; (void)ws_size;
  const float* x = (const float*)d_in[0];
  const int* esrc = (const int*)d_in[1];
  // d_in[2] edge_dst unused: dst(e) == e/16 by construction
  const float* gW1[3] = {(const float*)d_in[3], (const float*)d_in[9], (const float*)d_in[15]};
  const float* gb1[3] = {(const float*)d_in[4], (const float*)d_in[10], (const float*)d_in[16]};
  const float* gW2[3] = {(const float*)d_in[5], (const float*)d_in[11], (const float*)d_in[17]};
  const float* gb2[3] = {(const float*)d_in[6], (const float*)d_in[12], (const float*)d_in[18]};
  const float* gga[3] = {(const float*)d_in[7], (const float*)d_in[13], (const float*)d_in[19]};
  const float* gbe[3] = {(const float*)d_in[8], (const float*)d_in[14], (const float*)d_in[20]};
  const float* Wq = (const float*)d_in[21];
  const float* bq = (const float*)d_in[22];
  const float* Wk = (const float*)d_in[23];
  const float* bk = (const float*)d_in[24];
  const float* Wv = (const float*)d_in[25];
  const float* bv = (const float*)d_in[26];

  float* ws = (float*)d_ws;
  float* bufA = ws;
  float* bufB = bufA + (long)NT_TOTAL * HID;
  float* W1p = bufB + (long)NT_TOTAL * HID;
  float* stats = W1p + 64 * 64;
  float* fc = stats + 128;
  float* qb = fc + (long)N_BATCH * F_C_DIM;

  float* out = (float*)d_out;
  float* aout = out;                                       // (128,10,64)
  float* fout = out + (long)N_BATCH * N_AGENT * HID;       // (128,990,64)

  long elems = (long)NT_TOTAL * HID;
  pad_x_kernel<<<(int)((elems + 255) / 256), 256, 0, stream>>>(x, bufA);
  pad_w1_kernel<<<16, 256, 0, stream>>>(gW1[0], W1p);

  float* hcur = bufA;
  float* mbuf = bufB;
  for (int L = 0; L < 3; ++L) {
    agg_kernel<<<(NT_TOTAL * 32) / 256, 256, 0, stream>>>(hcur, esrc, mbuf);
    const float* W1use = (L == 0) ? (const float*)W1p : gW1[L];
    mlp_kernel<<<NT_TOTAL / 64, 128, 0, stream>>>(mbuf, W1use, gb1[L], gW2[L], gb2[L]);
    hipMemsetAsync(stats, 0, 128 * sizeof(float), stream);
    bn_stats_kernel<<<1024, 256, 0, stream>>>(mbuf, stats);
    bn_apply_kernel<<<(int)((elems + 255) / 256), 256, 0, stream>>>(mbuf, stats, gga[L], gbe[L]);
    float* t = hcur; hcur = mbuf; mbuf = t;
  }

  fc_build_kernel<<<N_BATCH, 256, 0, stream>>>(hcur, fc);
  q_kernel<<<(N_BATCH * N_AGENT * HID + 255) / 256, 256, 0, stream>>>(fc, Wq, bq, qb);
  long ftotal = (long)N_BATCH * N_FREE * 64;
  fcopy_kernel<<<(int)((ftotal + 255) / 256), 256, 0, stream>>>(hcur, fout);
  dim3 agrid(N_BATCH, N_AGENT);
  attn_kernel<<<agrid, 256, 0, stream>>>(hcur, qb, Wk, bk, Wv, bv, aout);
}